// BezierHungarianMatcher_88476326297800
// MI455X (gfx1250) — compile-verified
//
#include <hip/hip_runtime.h>

// Problem constants from the reference.
#define B_ 16
#define Q_ 512
#define T_ 128
#define C_ 4

typedef __attribute__((ext_vector_type(2))) float v2f;
typedef __attribute__((ext_vector_type(8))) float v8f;

// ---------------------------------------------------------------------------
// One workgroup (512 threads = 16 wave32s) per batch, one batch per WGP.
//
// Phase 1: build cost[t][q] = 5*L1(pos) + 1*(-prob[q][label[t]]) + 2*L1(dir)
//          16x16 tiles via V_WMMA_F32_16X16X4_F32 (A = -softmax 16x4,
//          B = one-hot labels 4x16, C = L1 terms), written straight into a
//          256 KB dynamic-LDS cost block (CDNA5: 320 KB LDS per workgroup).
//
// Phase 2: Jonker-Volgenant LAP (n=T rows, m=Q cols) mirroring the reference
//          lap_jv: LDS-resident cost rows, wave32 __shfl_xor argmin
//          butterflies (jnp.argmin tie-break), 4 barriers per inner step.
// ---------------------------------------------------------------------------
__global__ __launch_bounds__(Q_)
void matcher_kernel(const float* __restrict__ logits,
                    const float* __restrict__ pred_attr,
                    const int*   __restrict__ labels,
                    const float* __restrict__ tgt_attr,
                    long long*   __restrict__ out)
{
    const int   b    = blockIdx.x;
    const int   tid  = threadIdx.x;
    const int   lane = tid & 31;
    const int   wave = tid >> 5;          // 0..15
    const int   n = T_, m = Q_;
    const float BIGF = 1e9f;

    extern __shared__ float lcost[];      // [T_][Q_] fp32 = 256 KB dynamic LDS

    __shared__ float u[T_];
    __shared__ float v[Q_ + 1];
    __shared__ int   p[Q_ + 1];
    __shared__ float minv[Q_];
    __shared__ int   way[Q_];
    __shared__ int   used[Q_ + 1];
    __shared__ float wr_val[16];          // per-wave argmin partials
    __shared__ int   wr_idx[16];
    __shared__ int   pot[T_];             // pred index per target
    __shared__ int   s_j0, s_i0;
    __shared__ float s_delta;

    // ---------------- Phase 1: WMMA cost tiles -> LDS ----------------------
    {
        const int half = lane >> 4;       // 0: K={0,1}/M 0-7 ; 1: K={2,3}/M 8-15
        const int l16  = lane & 15;
        const int tTiles = T_ / 16, qTiles = Q_ / 16;       // 8 x 32 = 256 tiles
        for (int tile = wave; tile < tTiles * qTiles; tile += 16) {
            const int tIdx = tile % tTiles;
            const int qIdx = tile / tTiles;
            const int q0 = qIdx * 16, t0 = tIdx * 16;

            // A (16x4 f32): A[mq][k] = -softmax(logits[b][q0+mq])[k]
            const float4 lg = ((const float4*)logits)[(size_t)b * Q_ + q0 + l16];
            float mx  = fmaxf(fmaxf(lg.x, lg.y), fmaxf(lg.z, lg.w));
            float e0 = expf(lg.x - mx), e1 = expf(lg.y - mx);
            float e2 = expf(lg.z - mx), e3 = expf(lg.w - mx);
            float inv = 1.0f / (e0 + e1 + e2 + e3);
            v2f a;
            a.x = -(half ? e2 : e0) * inv;
            a.y = -(half ? e3 : e1) * inv;

            // B (4x16 f32): one-hot of labels (class gather as matmul)
            const int lbl = labels[b * T_ + t0 + l16];
            const int k0  = half * 2;
            v2f bm;
            bm.x = (lbl == k0    ) ? 1.0f : 0.0f;
            bm.y = (lbl == k0 + 1) ? 1.0f : 0.0f;

            // C accumulator: 5*L1(pos) + 2*L1(dir)
            const int    t  = t0 + l16;
            const float4 ta = ((const float4*)tgt_attr)[(size_t)b * T_ + t];
            v8f c;
#pragma unroll
            for (int i = 0; i < 8; ++i) {
                const int    mq = i + 8 * half;
                const float4 pa = ((const float4*)pred_attr)[(size_t)b * Q_ + q0 + mq];
                float pos = fabsf(pa.x - ta.x) + fabsf(pa.y - ta.y);
                float drc = fabsf(pa.z - ta.z) + fabsf(pa.w - ta.w);
                c[i] = 5.0f * pos + 2.0f * drc;
            }

            v8f d = __builtin_amdgcn_wmma_f32_16x16x4_f32(
                /*neg_a=*/false, a, /*neg_b=*/false, bm,
                /*c_mod=*/(short)0, c, /*reuse_a=*/false, /*reuse_b=*/false);

            // 8 consecutive q per lane -> two ds_store_b128 per lane
            float* orow = &lcost[(size_t)t * Q_ + q0 + 8 * half];
#pragma unroll
            for (int i = 0; i < 8; ++i) orow[i] = d[i];
        }
    }

    // ---------------- Phase 2: LAP init ------------------------------------
    for (int j = tid; j <= m; j += Q_) { v[j] = 0.0f; p[j] = -1; }
    if (tid < n) u[tid] = 0.0f;
    __syncthreads();                       // also covers Phase-1 LDS writes

    for (int i = 0; i < n; ++i) {
        minv[tid] = BIGF; way[tid] = 0; used[tid] = 0;
        if (tid == 0) { p[m] = i; s_j0 = m; used[m] = 0; }
        __syncthreads();

        while (true) {
            const int j0 = s_j0;
            if (p[j0] < 0) break;          // uniform across the block
            if (tid == 0) { used[j0] = 1; s_i0 = p[j0]; }
            __syncthreads();
            const int    i0   = s_i0;
            const float  ui0  = u[i0];
            const float* crow = &lcost[(size_t)i0 * Q_];   // LDS-resident row

            // reduced-cost update over unused columns
            float rv = BIGF;
            if (!used[tid]) {
                float cur = crow[tid] - ui0 - v[tid];
                if (cur < minv[tid]) { minv[tid] = cur; way[tid] = j0; }
                rv = minv[tid];
            }
            int ri = tid;

            // intra-wave lexicographic (val,idx) argmin butterfly: no barriers
#pragma unroll
            for (int off = 16; off > 0; off >>= 1) {
                float ov = __shfl_xor(rv, off, 32);
                int   oi = __shfl_xor(ri, off, 32);
                if (ov < rv || (ov == rv && oi < ri)) { rv = ov; ri = oi; }
            }
            if (lane == 0) { wr_val[wave] = rv; wr_idx[wave] = ri; }
            __syncthreads();

            // wave 0 reduces the 16 wave partials
            if (tid < 32) {
                float v2 = (lane < 16) ? wr_val[lane] : BIGF;
                int   i2 = (lane < 16) ? wr_idx[lane] : 0x7fffffff;
#pragma unroll
                for (int off = 16; off > 0; off >>= 1) {
                    float ov = __shfl_xor(v2, off, 32);
                    int   oi = __shfl_xor(i2, off, 32);
                    if (ov < v2 || (ov == v2 && oi < i2)) { v2 = ov; i2 = oi; }
                }
                if (lane == 0) { s_delta = v2; s_j0 = i2; }
            }
            __syncthreads();
            const float delta = s_delta;

            // dual updates: matched rows of used columns are distinct -> no atomics
            for (int jj = tid; jj <= m; jj += Q_) {
                if (used[jj]) { u[p[jj]] += delta; v[jj] -= delta; }
                else if (jj < m) minv[jj] -= delta;
            }
            __syncthreads();
        }

        if (tid == 0) {                    // augment along stored path
            int j0 = s_j0;
            while (j0 != m) { int j1 = way[j0]; p[j0] = p[j1]; j0 = j1; }
        }
        __syncthreads();
    }

    // p[0..m-1] = target per prediction column (-1 free); invert + argsort
    {
        int t = p[tid];
        if (t >= 0) pot[t] = tid;          // unique scatter
    }
    __syncthreads();
    if (tid < n) {
        const int myq = pot[tid];
        int r = 0;
        for (int t2 = 0; t2 < n; ++t2) r += (pot[t2] < myq) ? 1 : 0;
        out[(size_t)b * n + r]                   = (long long)myq;   // rows
        out[(size_t)B_ * T_ + (size_t)b * n + r] = (long long)tid;   // cols
    }
}

extern "C" void kernel_launch(void* const* d_in, const int* in_sizes, int n_in,
                              void* d_out, int out_size, void* d_ws, size_t ws_size,
                              hipStream_t stream) {
    const float* logits    = (const float*)d_in[0];   // [B,Q,C] f32
    const float* pred_attr = (const float*)d_in[1];   // [B,Q,4] f32
    const int*   labels    = (const int*)d_in[2];     // [B,T]   i32
    const float* tgt_attr  = (const float*)d_in[3];   // [B,T,4] f32
    long long*   out       = (long long*)d_out;       // rows[B,T] ++ cols[B,T]

    const size_t dynLds = (size_t)T_ * Q_ * sizeof(float);   // 256 KB
    hipFuncSetAttribute(reinterpret_cast<const void*>(matcher_kernel),
                        hipFuncAttributeMaxDynamicSharedMemorySize, (int)dynLds);
    matcher_kernel<<<B_, Q_, dynLds, stream>>>(logits, pred_attr, labels,
                                               tgt_attr, out);

    (void)in_sizes; (void)n_in; (void)out_size; (void)d_ws; (void)ws_size;
}